// Lin2d_43920335568980
// MI455X (gfx1250) — compile-verified
//
#include <hip/hip_runtime.h>

typedef __attribute__((ext_vector_type(2))) float v2f;
typedef __attribute__((ext_vector_type(8))) float v8f;

// ---------------------------------------------------------------------------
// Kernel 1: one thread computes P = linMt^N (2x2 fp32 chain, same order of
// magnitude of rounding as the reference's 200-step vector chain) and stores
// Q = P^T as [Q00,Q01,Q10,Q11] into workspace.  x' = x.P  =>
//   x0' = Q00*x0 + Q01*x1 ; x1' = Q10*x0 + Q11*x1
// ---------------------------------------------------------------------------
__global__ void lin2d_power_kernel(const float* __restrict__ linMt,
                                   const int* __restrict__ nPtr,
                                   float* __restrict__ q) {
    if (threadIdx.x != 0 || blockIdx.x != 0) return;
    const int N = nPtr[0];
    const float l00 = linMt[0], l01 = linMt[1];
    const float l10 = linMt[2], l11 = linMt[3];
    float p00 = 1.f, p01 = 0.f, p10 = 0.f, p11 = 1.f;
    for (int i = 0; i < N; ++i) {
        const float n00 = fmaf(p00, l00, p01 * l10);
        const float n01 = fmaf(p00, l01, p01 * l11);
        const float n10 = fmaf(p10, l00, p11 * l10);
        const float n11 = fmaf(p10, l01, p11 * l11);
        p00 = n00; p01 = n01; p10 = n10; p11 = n11;
    }
    q[0] = p00;  // Q00 = P[0][0]
    q[1] = p10;  // Q01 = P[1][0]
    q[2] = p01;  // Q10 = P[0][1]
    q[3] = p11;  // Q11 = P[1][1]
}

// ---------------------------------------------------------------------------
// Kernel 2: memory-bound pass.  Each wave transforms 32 states per iteration
// with a single V_WMMA_F32_16X16X4_F32.
//
// B (4x16) VGPR striping: VGPR0 = {K=0 lanes 0-15, K=2 lanes 16-31},
//                         VGPR1 = {K=1 lanes 0-15, K=3 lanes 16-31}.
// => b0 = x0[state==lane], b1 = x1[state==lane] for ALL 32 lanes (direct load).
//
// A (16x4) striping: VGPR0 = {K=0 lanes 0-15 (row=lane), K=2 lanes 16-31
// (row=lane-16)}, VGPR1 = {K=1, K=3}.  Coefficients:
//   rows 0,1 x cols 0,1 = Q-block for states 0-15  -> D rows 0,1
//   rows 8,9 x cols 2,3 = Q-block for states 16-31 -> D rows 8,9
// => A VGPR0: lane 0 -> Q00, lane 1 -> Q10, lane 24 (K=2,row8) -> Q00,
//    lane 25 -> Q10.  A VGPR1: lane 0 -> Q01, lane 1 -> Q11, lane 24 -> Q01,
//    lane 25 -> Q11.  Everything else 0.
//
// D (16x16) striping: VGPR0 = {row0 lanes 0-15, row8 lanes 16-31} = new x0 of
// state==lane; VGPR1 likewise new x1.  Direct float2 store.
// ---------------------------------------------------------------------------
__global__ void lin2d_apply_wmma(const float* __restrict__ xin_raw,
                                 const float* __restrict__ q,
                                 float* __restrict__ xout_raw,
                                 int nStates) {
    const int lane = threadIdx.x & 31;
    const unsigned wave   = (blockIdx.x * blockDim.x + threadIdx.x) >> 5;
    const unsigned nWaves = (gridDim.x * blockDim.x) >> 5;
    const unsigned nGroups = (unsigned)nStates >> 5;   // full groups of 32

    const float Q00 = q[0], Q01 = q[1], Q10 = q[2], Q11 = q[3];

    v2f a; a.x = 0.0f; a.y = 0.0f;
    if (lane == 0 || lane == 24) { a.x = Q00; a.y = Q01; }
    if (lane == 1 || lane == 25) { a.x = Q10; a.y = Q11; }

    const float2* __restrict__ xin  = (const float2*)xin_raw;
    float2*       __restrict__ xout = (float2*)xout_raw;

    for (unsigned g = wave; g < nGroups; g += nWaves) {
        const unsigned idx = (g << 5) + (unsigned)lane;
        const float2 v = xin[idx];
        v2f b; b.x = v.x; b.y = v.y;
        v8f c = {};
        c = __builtin_amdgcn_wmma_f32_16x16x4_f32(
                /*neg_a=*/false, a, /*neg_b=*/false, b,
                /*c_mod=*/(short)0, c, /*reuse_a=*/false, /*reuse_b=*/false);
        float2 r; r.x = c[0]; r.y = c[1];
        xout[idx] = r;
    }

    // Scalar tail for nStates % 32 (no WMMA here; EXEC may diverge freely).
    const int tid   = blockIdx.x * blockDim.x + threadIdx.x;
    const int total = gridDim.x * blockDim.x;
    for (int i = (int)(nGroups << 5) + tid; i < nStates; i += total) {
        const float2 v = xin[i];
        float2 r;
        r.x = fmaf(v.x, Q00, v.y * Q01);
        r.y = fmaf(v.x, Q10, v.y * Q11);
        xout[i] = r;
    }
}

extern "C" void kernel_launch(void* const* d_in, const int* in_sizes, int n_in,
                              void* d_out, int out_size, void* d_ws, size_t ws_size,
                              hipStream_t stream) {
    const float* x     = (const float*)d_in[0];
    const float* linMt = (const float*)d_in[1];
    const int*   nPtr  = (const int*)d_in[2];
    float*       out   = (float*)d_out;
    float*       q     = (float*)d_ws;

    const int nStates = in_sizes[0] / 2;

    // Tiny sequential kernel: compute linMt^N once.
    lin2d_power_kernel<<<1, 32, 0, stream>>>(linMt, nPtr, q);

    // Streaming pass: 8 waves/block, target ~8 groups (256 states) per wave.
    const int nGroups = nStates >> 5;
    int blocks = (nGroups + 63) / 64;
    if (blocks < 1) blocks = 1;
    if (blocks > 8192) blocks = 8192;
    lin2d_apply_wmma<<<blocks, 256, 0, stream>>>(x, q, out, nStates);
}